// GCNVar_29231547416620
// MI455X (gfx1250) — compile-verified
//
#include <hip/hip_runtime.h>
#include <hip/hip_bf16.h>

#define NN    100000
#define NE    1200000
#define INF_  128
#define HID   64
#define NCLS  40

typedef __attribute__((ext_vector_type(2))) float v2f;
typedef __attribute__((ext_vector_type(8))) float v8f;

// ---------------- degree / normalization ----------------

__global__ void k_deg_init(float* __restrict__ deg) {
    int i = blockIdx.x * blockDim.x + threadIdx.x;
    if (i < NN) deg[i] = 1.0f;   // self-loop
}

__global__ void k_deg_count(const int* __restrict__ dst, float* __restrict__ deg) {
    int i = blockIdx.x * blockDim.x + threadIdx.x;
    if (i < NE)
        __hip_atomic_fetch_add(&deg[dst[i]], 1.0f, __ATOMIC_RELAXED, __HIP_MEMORY_SCOPE_AGENT);
}

__global__ void k_dinv(float* __restrict__ deg) {
    int i = blockIdx.x * blockDim.x + threadIdx.x;
    if (i < NN) deg[i] = rsqrtf(deg[i]);   // deg >= 1 always (self-loop)
}

// ---------------- WMMA GEMM: D[M x 64] = A[M x K] @ B[K x 64] ----------------
// One 16x16 output tile per wave; 4 waves/block cover the 64 output columns.
// A-frag (16x4 f32): lane m = lane&15, half = lane>>4 selects K pair {2h, 2h+1}.
// D-frag: VGPR v -> M = v + 8*half, N = lane&15.

__global__ __launch_bounds__(128)
void k_gemm_wmma64(const float* __restrict__ A, const float* __restrict__ B,
                   float* __restrict__ D, int K) {
    const int lane = threadIdx.x & 31;
    const int wave = threadIdx.x >> 5;         // column tile 0..3
    const int m    = lane & 15;
    const int half = lane >> 4;
    const int row0 = blockIdx.x * 16;
    const int n0   = wave * 16;

    const float* __restrict__ Arow = A + (size_t)(row0 + m) * K;

    v8f acc = {};
    for (int k = 0; k < K; k += 4) {
        const int ka = k + 2 * half;
        v2f a, b;
        a.x = Arow[ka];
        a.y = Arow[ka + 1];
        b.x = B[(size_t)ka       * HID + n0 + m];
        b.y = B[(size_t)(ka + 1) * HID + n0 + m];
        acc = __builtin_amdgcn_wmma_f32_16x16x4_f32(
            /*neg_a=*/false, a, /*neg_b=*/false, b,
            /*c_mod=*/(short)0, acc, /*reuse_a=*/false, /*reuse_b=*/false);
    }

    float* __restrict__ Dtile = D + (size_t)row0 * HID + n0;
#pragma unroll
    for (int v = 0; v < 8; ++v) {
        const int mo = v + 8 * half;
        Dtile[(size_t)mo * HID + m] = acc[v];
    }
}

// ---------------- propagate: out = scatter_add(norm * h[src]) ----------------

// out[i][f] = h[i][f] * dinv[i]^2   (self-loop term, also zero-initializes out)
__global__ void k_self_init(const float* __restrict__ h, const float* __restrict__ dinv,
                            float* __restrict__ out) {
    int i = blockIdx.x * blockDim.x + threadIdx.x;
    if (i < NN * HID) {
        const int node = i >> 6;
        const float d = dinv[node];
        out[i] = h[i] * d * d;
    }
}

// one wave32 per edge, 2 features per lane
__global__ __launch_bounds__(256)
void k_scatter(const float* __restrict__ h, const float* __restrict__ dinv,
               const int* __restrict__ src, const int* __restrict__ dst,
               float* __restrict__ out) {
    const int e = (int)((blockIdx.x * (unsigned)blockDim.x + threadIdx.x) >> 5);
    if (e >= NE) return;
    const int lane = threadIdx.x & 31;
    const int s = src[e];
    const int d = dst[e];
    const float w = dinv[s] * dinv[d];
    const size_t sb = (size_t)s * HID;
    const size_t db = (size_t)d * HID;
    const float v0 = h[sb + lane]      * w;
    const float v1 = h[sb + lane + 32] * w;
    __hip_atomic_fetch_add(&out[db + lane],      v0, __ATOMIC_RELAXED, __HIP_MEMORY_SCOPE_AGENT);
    __hip_atomic_fetch_add(&out[db + lane + 32], v1, __ATOMIC_RELAXED, __HIP_MEMORY_SCOPE_AGENT);
}

__global__ void k_bias_relu(float* __restrict__ io, const float* __restrict__ b) {
    int i = blockIdx.x * blockDim.x + threadIdx.x;
    if (i < NN * HID) io[i] = fmaxf(io[i] + b[i & 63], 0.0f);
}

__global__ void k_bias(float* __restrict__ io, const float* __restrict__ b) {
    int i = blockIdx.x * blockDim.x + threadIdx.x;
    if (i < NN * HID) io[i] += b[i & 63];
}

// ---------------- classifier head + softmax/argmax ----------------

__global__ void k_head(const float* __restrict__ emb, const float* __restrict__ Wc,
                       const float* __restrict__ bc, float* __restrict__ logits) {
    int i = blockIdx.x * blockDim.x + threadIdx.x;
    if (i >= NN * NCLS) return;
    const int node = i / NCLS;
    const int c    = i - node * NCLS;
    const float* __restrict__ er = emb + (size_t)node * HID;
    float s = bc[c];
#pragma unroll 8
    for (int k = 0; k < HID; ++k)
        s = fmaf(er[k], Wc[(size_t)k * NCLS + c], s);
    logits[i] = s;
}

__global__ void k_softmax_argmax(const float* __restrict__ logits,
                                 float* __restrict__ soft, float* __restrict__ hard) {
    int node = blockIdx.x * blockDim.x + threadIdx.x;
    if (node >= NN) return;
    const float* __restrict__ l = logits + (size_t)node * NCLS;
    float m = l[0];
    int best = 0;
#pragma unroll
    for (int c = 1; c < NCLS; ++c) {
        const float v = l[c];
        if (v > m) { m = v; best = c; }   // first-index tie-break == jnp.argmax
    }
    float ssum = 0.0f;
    float e[NCLS];
#pragma unroll
    for (int c = 0; c < NCLS; ++c) { e[c] = __expf(l[c] - m); ssum += e[c]; }
    const float inv = 1.0f / ssum;
    float* __restrict__ so = soft + (size_t)node * NCLS;
#pragma unroll
    for (int c = 0; c < NCLS; ++c) so[c] = e[c] * inv;
    hard[node] = (float)best;
}

// ---------------- driver ----------------

extern "C" void kernel_launch(void* const* d_in, const int* in_sizes, int n_in,
                              void* d_out, int out_size, void* d_ws, size_t ws_size,
                              hipStream_t stream) {
    const float* x   = (const float*)d_in[0];
    const int*   ei  = (const int*)  d_in[1];     // [2, E] flat: src row then dst row
    const float* W1  = (const float*)d_in[2];
    const float* b1  = (const float*)d_in[3];
    const float* W2  = (const float*)d_in[4];
    const float* b2  = (const float*)d_in[5];
    const float* Wc  = (const float*)d_in[6];
    const float* bc  = (const float*)d_in[7];

    const int* src = ei;
    const int* dst = ei + NE;

    float* out = (float*)d_out;
    float* logits = out;                                    // [N, 40]
    float* emb    = out + (size_t)NN * NCLS;                // [N, 64]
    float* soft   = out + (size_t)NN * (NCLS + HID);        // [N, 40]
    float* hard   = out + (size_t)NN * (NCLS + HID + NCLS); // [N]

    float* ws   = (float*)d_ws;
    float* dinv = ws;                                       // [N]
    float* bufA = ws + NN;                                  // [N, 64]
    float* bufB = bufA + (size_t)NN * HID;                  // [N, 64]

    const int T = 256;
    const int gN    = (NN + T - 1) / T;
    const int gE    = (NE + T - 1) / T;
    const int gNH   = (NN * HID + T - 1) / T;
    const int gNC   = (NN * NCLS + T - 1) / T;
    const int gEdgeWaves = (int)(((size_t)NE * 32 + T - 1) / T);
    const int gGemm = NN / 16;   // 6250, exact

    // normalization: deg -> dinv (in place)
    k_deg_init <<<gN, T, 0, stream>>>(dinv);
    k_deg_count<<<gE, T, 0, stream>>>(dst, dinv);
    k_dinv     <<<gN, T, 0, stream>>>(dinv);

    // layer 1: h1 = x @ W1 ; propagate ; relu(+b1)
    k_gemm_wmma64<<<gGemm, 128, 0, stream>>>(x, W1, bufA, INF_);
    k_self_init  <<<gNH, T, 0, stream>>>(bufA, dinv, bufB);
    k_scatter    <<<gEdgeWaves, T, 0, stream>>>(bufA, dinv, src, dst, bufB);
    k_bias_relu  <<<gNH, T, 0, stream>>>(bufB, b1);

    // layer 2: h2 = h @ W2 ; propagate into d_out embedding ; +b2
    k_gemm_wmma64<<<gGemm, 128, 0, stream>>>(bufB, W2, bufA, HID);
    k_self_init  <<<gNH, T, 0, stream>>>(bufA, dinv, emb);
    k_scatter    <<<gEdgeWaves, T, 0, stream>>>(bufA, dinv, src, dst, emb);
    k_bias       <<<gNH, T, 0, stream>>>(emb, b2);

    // head + softmax/argmax
    k_head          <<<gNC, T, 0, stream>>>(emb, Wc, bc, logits);
    k_softmax_argmax<<<gN,  T, 0, stream>>>(logits, soft, hard);
}